// NonMaximaSuppression2d_3513283248649
// MI455X (gfx1250) — compile-verified
//
#include <hip/hip_runtime.h>

// NMS2d 3x3, replicate pad, center-excluded window max clamped at 0.
// Memory-bound stencil (~570 MB HBM traffic, ~0.7 GFLOP -> pure bandwidth).
// Rows are staged into a 4-deep LDS ring via CDNA5 async global->LDS DMA,
// software-pipelined one row ahead (s_wait_asynccnt <= 1), with a 3-row
// sliding window held in registers and b128 loads/stores everywhere.

#define WD   512   // plane width (floats)
#define TILE 32    // output rows per block
#define TPB  128   // threads per block; 128 * 4 floats = 512 = full row
#define NBUF 4     // LDS row ring depth (pipelined by one row)

#if defined(__has_builtin)
#if __has_builtin(__builtin_amdgcn_global_load_async_to_lds_b128)
#define HAVE_ASYNC_LDS 1
#endif
#if __has_builtin(__builtin_amdgcn_s_wait_asynccnt)
#define HAVE_WAIT_ASYNC 1
#endif
#endif

// Builtin takes pointers to 16-byte int vectors (per hipcc diagnostic).
typedef int v4i __attribute__((ext_vector_type(4)));
typedef __attribute__((address_space(1))) v4i gv4i_t;  // global
typedef __attribute__((address_space(3))) v4i lv4i_t;  // LDS

static __device__ __forceinline__ void wait_async_le1() {
#if defined(HAVE_WAIT_ASYNC)
  __builtin_amdgcn_s_wait_asynccnt(1);   // all but most recent DMA complete
#elif defined(HAVE_ASYNC_LDS)
  asm volatile("s_wait_asynccnt 1" ::: "memory");
#endif
}

struct Row {
  float4 c;   // raw values
  float4 h3;  // 3-wide horizontal max (incl. center)
  float4 lr;  // left/right max (excl. center)
};

__global__ __launch_bounds__(TPB) void nms2d_kernel(const float* __restrict__ xin,
                                                    float* __restrict__ out,
                                                    int H) {
  __shared__ float lds[NBUF][WD];

  const int t  = threadIdx.x;   // 0..127
  const int x0 = t << 2;        // 0,4,...,508
  const size_t plane = blockIdx.y;
  const int r0 = blockIdx.x * TILE;

  const float* __restrict__ px = xin + plane * (size_t)H * WD;
  float* __restrict__ po       = out + plane * (size_t)H * WD;

  // Stage one (row-clamped) 512-float row into LDS ring slot `buf`.
  auto stage = [&](int row, int buf) {
    row = row < 0 ? 0 : (row >= H ? H - 1 : row);
    const float* src = px + (size_t)row * WD + x0;
#if defined(HAVE_ASYNC_LDS)
    __builtin_amdgcn_global_load_async_to_lds_b128(
        (gv4i_t*)(void*)const_cast<float*>(src),
        (lv4i_t*)(void*)&lds[buf][x0],
        /*offset=*/0, /*cpol=*/0);
#else
    *(float4*)&lds[buf][x0] = *(const float4*)src;
#endif
  };

  // Read a staged row from LDS; compute raw / h3 / lr with replicate edges.
  auto readrow = [&](int buf) {
    Row v;
    float4 c = *(const float4*)&lds[buf][x0];
    // replicate pad: clamped neighbor equals the element itself at edges
    float l = (x0 == 0)       ? c.x : lds[buf][x0 - 1];
    float r = (x0 + 4 == WD)  ? c.w : lds[buf][x0 + 4];
    v.c = c;
    v.h3.x = fmaxf(fmaxf(l,   c.x), c.y);
    v.h3.y = fmaxf(fmaxf(c.x, c.y), c.z);
    v.h3.z = fmaxf(fmaxf(c.y, c.z), c.w);
    v.h3.w = fmaxf(fmaxf(c.z, c.w), r);
    v.lr.x = fmaxf(l,   c.y);
    v.lr.y = fmaxf(c.x, c.z);
    v.lr.z = fmaxf(c.y, c.w);
    v.lr.w = fmaxf(c.z, r);
    return v;
  };

  // Prologue: stage rows r0-1, r0, r0+1 into slots 0,1,2 (r0+1 stays in
  // flight); wait for the first two, then pull them into registers.
  stage(r0 - 1, 0);
  stage(r0,     1);
  stage(r0 + 1, 2);
  wait_async_le1();
  __syncthreads();
  Row A = readrow(0);  // row r-1
  Row B = readrow(1);  // row r

  for (int k = 0; k < TILE; ++k) {
    const int r = r0 + k;
    stage(r + 2, (k + 3) & (NBUF - 1));   // prefetch row r+2 (overlaps compute)
    wait_async_le1();                     // row r+1 DMA complete (in-order)
    __syncthreads();
    Row Cv = readrow((k + 2) & (NBUF - 1));  // row r+1

    const float4 raw = B.c;
    float4 o;
    {
      float m = fmaxf(fmaxf(A.h3.x, Cv.h3.x), fmaxf(B.lr.x, 0.0f));
      o.x = (raw.x > m) ? raw.x : 0.0f;
      m = fmaxf(fmaxf(A.h3.y, Cv.h3.y), fmaxf(B.lr.y, 0.0f));
      o.y = (raw.y > m) ? raw.y : 0.0f;
      m = fmaxf(fmaxf(A.h3.z, Cv.h3.z), fmaxf(B.lr.z, 0.0f));
      o.z = (raw.z > m) ? raw.z : 0.0f;
      m = fmaxf(fmaxf(A.h3.w, Cv.h3.w), fmaxf(B.lr.w, 0.0f));
      o.w = (raw.w > m) ? raw.w : 0.0f;
    }
    *(float4*)(po + (size_t)r * WD + x0) = o;

    A = B;
    B = Cv;
  }
  // One DMA (row r0+TILE+1, clamped in-bounds) may still be in flight into
  // LDS at exit; S_ENDPGM performs an implicit wait-idle.
}

extern "C" void kernel_launch(void* const* d_in, const int* in_sizes, int n_in,
                              void* d_out, int out_size, void* d_ws, size_t ws_size,
                              hipStream_t stream) {
  (void)n_in; (void)out_size; (void)d_ws; (void)ws_size;
  const float* xin = (const float*)d_in[0];
  float* out = (float*)d_out;

  const int HW = WD * WD;                       // 512*512 plane
  const int planes = in_sizes[0] / HW;          // 16*17 = 272
  const int H = WD;                             // 512

  dim3 grid(H / TILE, planes);                  // (16, 272)
  dim3 block(TPB);
  nms2d_kernel<<<grid, block, 0, stream>>>(xin, out, H);
}